// GCNLayer_11828339933792
// MI455X (gfx1250) — compile-verified
//
#include <hip/hip_runtime.h>
#include <hip/hip_bf16.h>

typedef __attribute__((ext_vector_type(2))) float v2f;
typedef __attribute__((ext_vector_type(8))) float v8f;

#define IN_F 256
#define OUT_F 32

// ---------------------------------------------------------------------------
// Zero workspace degree arrays and the output accumulator (must run every
// call: harness does not re-poison between graph replays).
// ---------------------------------------------------------------------------
__global__ void gcn_init(int* __restrict__ outdeg, int* __restrict__ indeg,
                         float* __restrict__ out, int N) {
  const int stride = gridDim.x * blockDim.x;
  const int tid = blockIdx.x * blockDim.x + threadIdx.x;
  const int totalOut = N * OUT_F;
  for (int i = tid; i < totalOut; i += stride) out[i] = 0.0f;
  for (int i = tid; i < N; i += stride) { outdeg[i] = 0; indeg[i] = 0; }
}

// ---------------------------------------------------------------------------
// Degree histograms via integer atomics (L2-resident, 400KB each).
// ---------------------------------------------------------------------------
__global__ void gcn_degree(const int* __restrict__ src, const int* __restrict__ dst,
                           int* __restrict__ outdeg, int* __restrict__ indeg, int E) {
  const int stride = gridDim.x * blockDim.x;
  for (int e = blockIdx.x * blockDim.x + threadIdx.x; e < E; e += stride) {
    atomicAdd(&outdeg[src[e]], 1);
    atomicAdd(&indeg[dst[e]], 1);
  }
}

// ---------------------------------------------------------------------------
// h = (feat * outdeg^-0.5) @ W   using V_WMMA_F32_16X16X4_F32 (exact f32).
// One wave32 computes a 16-row x 32-col tile: two 16x16 f32 accumulators,
// K swept 4 at a time (64 WMMA pairs).
//
// A 16x4 f32 layout (ISA 7.12.2): lane L -> M = L%16;
//   VGPR0 = K0+2*(L/16), VGPR1 = K0+2*(L/16)+1  (contiguous pair -> one b64 load)
// B 4x16 f32 (row striped across lanes, half-wave K split):
//   lane L -> N = L%16; VGPR0 = W[K0+2*(L/16)][N], VGPR1 = W[K0+2*(L/16)+1][N]
// C/D 16x16 f32: VGPR r -> row (r + 8*(L/16)), col L%16.
// ---------------------------------------------------------------------------
__global__ void gcn_gemm_wmma(const float* __restrict__ feat,
                              const float* __restrict__ weight,
                              const int* __restrict__ outdeg,
                              float* __restrict__ h, int N) {
  const int lane = threadIdx.x & 31;
  const int wavesPerBlock = blockDim.x >> 5;
  const int tile = blockIdx.x * wavesPerBlock + (threadIdx.x >> 5);
  const int numTiles = N >> 4;                 // N is a multiple of 16 (100000)
  if (tile >= numTiles) return;                // wave-uniform: EXEC stays all-1s

  const int m  = (tile << 4) + (lane & 15);    // this lane's A row
  const int n  = lane & 15;                    // this lane's B/C column
  const int hi = lane >> 4;                    // half-wave K offset selector

  int d = outdeg[m];
  const float s = rsqrtf((float)(d < 1 ? 1 : d));   // norm='both' source scale

  const float* __restrict__ arow = feat + (size_t)m * IN_F;

  v8f acc0 = {};   // columns 0..15
  v8f acc1 = {};   // columns 16..31

#pragma unroll 4
  for (int k0 = 0; k0 < IN_F; k0 += 4) {
    const int ka = k0 + 2 * hi;
    v2f a;
    a.x = arow[ka] * s;
    a.y = arow[ka + 1] * s;

    v2f b0, b1;
    b0.x = weight[ka * OUT_F + n];
    b0.y = weight[(ka + 1) * OUT_F + n];
    b1.x = weight[ka * OUT_F + 16 + n];
    b1.y = weight[(ka + 1) * OUT_F + 16 + n];

    // 8 args: (neg_a, A, neg_b, B, c_mod, C, reuse_a, reuse_b)
    acc0 = __builtin_amdgcn_wmma_f32_16x16x4_f32(false, a, false, b0,
                                                 (short)0, acc0, false, false);
    acc1 = __builtin_amdgcn_wmma_f32_16x16x4_f32(false, a, false, b1,
                                                 (short)0, acc1, false, false);
  }

  // Write back D per the 16x16 f32 C/D VGPR layout.
#pragma unroll
  for (int r = 0; r < 8; ++r) {
    const int row = (tile << 4) + r + 8 * hi;
    h[(size_t)row * OUT_F + n]      = acc0[r];
    h[(size_t)row * OUT_F + 16 + n] = acc1[r];
  }
}

// ---------------------------------------------------------------------------
// Edge scatter: out[dst] += h[src].  One wave per edge iteration, lane =
// column -> each edge is ONE coalesced 128B load + 32 contiguous
// global_atomic_add_f32 (h and out are both L2-resident: 12.8 MB each).
// ---------------------------------------------------------------------------
__global__ void gcn_scatter(const float* __restrict__ h,
                            const int* __restrict__ src,
                            const int* __restrict__ dst,
                            float* __restrict__ out, int E) {
  const int lane = threadIdx.x & 31;
  const int wave = (blockIdx.x * blockDim.x + threadIdx.x) >> 5;
  const int numWaves = (gridDim.x * blockDim.x) >> 5;
  for (int e = wave; e < E; e += numWaves) {
    const int sn = src[e];                     // wave-uniform -> scalar load
    const int dn = dst[e];
    const float v = h[(size_t)sn * OUT_F + lane];
    atomicAdd(&out[(size_t)dn * OUT_F + lane], v);
  }
}

// ---------------------------------------------------------------------------
// out = relu(out * indeg^-0.5 + bias)
// ---------------------------------------------------------------------------
__global__ void gcn_finalize(float* __restrict__ out, const int* __restrict__ indeg,
                             const float* __restrict__ bias, int N) {
  const int i = blockIdx.x * blockDim.x + threadIdx.x;
  const int total = N * OUT_F;
  if (i >= total) return;
  const int node = i >> 5;          // OUT_F == 32
  const int col  = i & 31;
  int d = indeg[node];
  const float s = rsqrtf((float)(d < 1 ? 1 : d));
  float v = out[i] * s + bias[col];
  out[i] = v > 0.0f ? v : 0.0f;
}

extern "C" void kernel_launch(void* const* d_in, const int* in_sizes, int n_in,
                              void* d_out, int out_size, void* d_ws, size_t ws_size,
                              hipStream_t stream) {
  const float* feat   = (const float*)d_in[0];
  const int*   src    = (const int*)d_in[1];
  const int*   dst    = (const int*)d_in[2];
  const float* weight = (const float*)d_in[3];
  const float* bias   = (const float*)d_in[4];
  float* out = (float*)d_out;

  const int N = in_sizes[0] / IN_F;   // 100000
  const int E = in_sizes[1];          // 1600000

  // Workspace layout: [outdeg: N int][indeg: N int][h: N*32 float]
  char* ws = (char*)d_ws;
  int*   outdeg = (int*)ws;
  int*   indeg  = (int*)(ws + (size_t)N * sizeof(int));
  float* h      = (float*)(ws + (size_t)2 * N * sizeof(int));

  const int TPB = 256;

  // 1. zero degrees + output accumulator
  gcn_init<<<4096, TPB, 0, stream>>>(outdeg, indeg, out, N);

  // 2. degree histograms
  gcn_degree<<<4096, TPB, 0, stream>>>(src, dst, outdeg, indeg, E);

  // 3. WMMA GEMM: h = (feat * outdeg^-0.5) @ W
  const int numTiles = N >> 4;                       // 6250 (N % 16 == 0)
  const int wavesPerBlock = TPB / 32;                // 8
  const int gemmBlocks = (numTiles + wavesPerBlock - 1) / wavesPerBlock;
  gcn_gemm_wmma<<<gemmBlocks, TPB, 0, stream>>>(feat, weight, outdeg, h, N);

  // 4. edge scatter (wave-per-edge, lane-per-column)
  gcn_scatter<<<8192, TPB, 0, stream>>>(h, src, dst, out, E);

  // 5. normalize + bias + relu
  gcn_finalize<<<(N * OUT_F + TPB - 1) / TPB, TPB, 0, stream>>>(out, indeg, bias, N);
}